// PipeLine_55697135895330
// MI455X (gfx1250) — compile-verified
//
#include <hip/hip_runtime.h>
#include <math.h>

// Problem constants (from the reference).
#define B_  2
#define N_  4096
#define E_  4
#define D_  64
#define AD_ 32
#define NE_ (N_ * E_)   // 16384

typedef float v2f __attribute__((ext_vector_type(2)));
typedef float v8f __attribute__((ext_vector_type(8)));

// D(16x16,f32) = A(16x4,f32) * B(4x16,f32) + C  — CDNA5 wave32 WMMA.
// A frag: lane l (l<16): row=l, v0=K0,v1=K1 ; (l>=16): row=l-16, v0=K2,v1=K3.
// B frag: lane l (l<16): col=l, v0=K0,v1=K1 ; (l>=16): col=l-16, v0=K2,v1=K3.
// C/D: vgpr r, lanes0-15: (M=r, N=lane); lanes16-31: (M=r+8, N=lane-16).
static __device__ __forceinline__ v8f wmma_f32(v2f a, v2f b, v8f c) {
  return __builtin_amdgcn_wmma_f32_16x16x4_f32(
      /*neg_a=*/false, a, /*neg_b=*/false, b,
      /*c_mod=*/(short)0, c, /*reuse_a=*/false, /*reuse_b=*/false);
}

static __device__ __forceinline__ v8f vzero8() {
  v8f z = {0.f, 0.f, 0.f, 0.f, 0.f, 0.f, 0.f, 0.f};
  return z;
}

// ---------------------------------------------------------------------------
// Kernel 1: in_flat_T[b][o][e*N+n] = sum_d prop[b][n][d] * W_in[e][o][d] + b_in[e][o]
// Stored TRANSPOSED ([b][o][m]) so the big GEMM's B-fragments are contiguous
// 8-byte loads along K. One wave per 16-row x 64-col tile.
// ---------------------------------------------------------------------------
__global__ __launch_bounds__(256) void k_in_transform(
    const float* __restrict__ prop, const float* __restrict__ Win,
    const float* __restrict__ bin, float* __restrict__ in_flat_T) {
  const int tid  = threadIdx.x;
  const int lane = tid & 31;
  const int warp = tid >> 5;
  const int tile = blockIdx.x * 8 + warp;   // B_*E_*(N_/16) = 2048 tiles
  const int b    = tile >> 10;              // / (E_*256)
  const int rem  = tile & 1023;
  const int e    = rem >> 8;
  const int n0   = (rem & 255) << 4;
  const int col  = lane & 15;
  const int kh   = lane >> 4;

  v8f c[4] = {vzero8(), vzero8(), vzero8(), vzero8()};

  const float* ap = prop + ((size_t)b * N_ + n0 + col) * D_ + kh * 2;
  const float* wp = Win + (size_t)e * D_ * D_ + kh * 2;

  #pragma unroll 4
  for (int d0 = 0; d0 < D_; d0 += 4) {
    v2f a = *(const v2f*)(ap + d0);
    #pragma unroll
    for (int j = 0; j < 4; ++j) {
      v2f bb = *(const v2f*)(wp + (size_t)(j * 16 + col) * D_ + d0);
      c[j] = wmma_f32(a, bb, c[j]);
    }
  }

  const size_t obase   = (size_t)b * D_ * NE_;
  const int    rowbase = e * N_ + n0 + kh * 8;
  #pragma unroll
  for (int j = 0; j < 4; ++j) {
    const int   o    = j * 16 + col;
    const float bias = bin[e * D_ + o];
    #pragma unroll
    for (int v = 0; v < 8; ++v)
      in_flat_T[obase + (size_t)o * NE_ + rowbase + v] = c[j][v] + bias;
  }
}

// ---------------------------------------------------------------------------
// Kernel 2: a_in[b][n][o] = sum_m A[b][n][m] * in_flat_T[b][o][m]
// HBM-bound on the 536 MB A stream (each element read exactly once);
// in_flat_T (8 MB) is L2-resident and reused by every block.
// Register blocking: each wave owns a 32-row x 64-col tile (8 x v8f accums),
// so a K-step of 4 costs 6 b64 loads for 8 WMMAs (B-frags reused in regs).
// One block (8 waves) per 32-row tile, 8-way split-K, LDS ds_add_f32 reduce.
// ---------------------------------------------------------------------------
__global__ __launch_bounds__(256) void k_agg(
    const float* __restrict__ Amat, const float* __restrict__ in_flat_T,
    float* __restrict__ a_in) {
  __shared__ float red[32 * 64];
  const int tid  = threadIdx.x;
  const int lane = tid & 31;
  const int warp = tid >> 5;
  const int b    = blockIdx.x >> 7;          // grid = B_ * 128
  const int n0   = (blockIdx.x & 127) << 5;  // 32-row tile
  const int col  = lane & 15;
  const int kh   = lane >> 4;

  for (int i = tid; i < 32 * 64; i += 256) red[i] = 0.f;
  __syncthreads();

  v8f c0[4] = {vzero8(), vzero8(), vzero8(), vzero8()};  // rows n0 .. n0+15
  v8f c1[4] = {vzero8(), vzero8(), vzero8(), vzero8()};  // rows n0+16 .. n0+31

  const float* ap0 = Amat + ((size_t)b * N_ + n0 + col) * NE_ + kh * 2;
  const float* ap1 = ap0 + (size_t)16 * NE_;
  const float* bp  = in_flat_T + (size_t)b * D_ * NE_ + kh * 2;

  const int m_beg = warp * (NE_ / 8);
  const int m_end = m_beg + (NE_ / 8);
  #pragma unroll 2
  for (int m0 = m_beg; m0 < m_end; m0 += 4) {
    v2f a0 = *(const v2f*)(ap0 + m0);
    v2f a1 = *(const v2f*)(ap1 + m0);
    __builtin_prefetch(ap0 + m0 + 512, 0, 0);  // ~2KB ahead in each lane's A stream
    __builtin_prefetch(ap1 + m0 + 512, 0, 0);
    #pragma unroll
    for (int j = 0; j < 4; ++j) {
      v2f bb = *(const v2f*)(bp + (size_t)(j * 16 + col) * NE_ + m0);
      c0[j] = wmma_f32(a0, bb, c0[j]);
      c1[j] = wmma_f32(a1, bb, c1[j]);
    }
  }

  // Cross-wave split-K reduction via LDS float atomics (ds_add_f32).
  #pragma unroll
  for (int j = 0; j < 4; ++j) {
    #pragma unroll
    for (int v = 0; v < 8; ++v) {
      atomicAdd(&red[(v + kh * 8) * 64 + j * 16 + col], c0[j][v]);
      atomicAdd(&red[(v + kh * 8 + 16) * 64 + j * 16 + col], c1[j][v]);
    }
  }
  __syncthreads();

  for (int i = tid; i < 32 * 64; i += 256) {
    const int r = i >> 6, o = i & 63;
    a_in[((size_t)b * N_ + n0 + r) * D_ + o] = red[i];
  }
}

// ---------------------------------------------------------------------------
// Kernel 3: GRU update. One wave per 16-row tile.
//   r = sigmoid([a_in,s] Wr^T + br), z = sigmoid([a_in,s] Wz^T + bz)
//   h = tanh([a_in, r*s] Wh^T + bh), s' = (1-z)*s + z*h
// r*s is bounced through LDS to convert C-layout -> A-fragment layout.
// ---------------------------------------------------------------------------
__global__ __launch_bounds__(32) void k_gru(
    const float* __restrict__ a_in, const float* __restrict__ prop,
    const float* __restrict__ Wr, const float* __restrict__ br,
    const float* __restrict__ Wz, const float* __restrict__ bz,
    const float* __restrict__ Wh, const float* __restrict__ bh,
    float* __restrict__ new_state) {
  __shared__ float rp[16 * 64];
  const int lane = threadIdx.x & 31;
  const int b    = blockIdx.x >> 8;
  const int n0   = (blockIdx.x & 255) << 4;
  const int col  = lane & 15;
  const int kh   = lane >> 4;

  const float* a0 = a_in + ((size_t)b * N_ + n0 + col) * D_ + kh * 2;
  const float* a1 = prop + ((size_t)b * N_ + n0 + col) * D_ + kh * 2;

  // --- r and z pre-activations (shared A fragments), K = 128 ---
  v8f cr[4] = {vzero8(), vzero8(), vzero8(), vzero8()};
  v8f cz[4] = {vzero8(), vzero8(), vzero8(), vzero8()};
  #pragma unroll 2
  for (int k0 = 0; k0 < 2 * D_; k0 += 4) {
    v2f a = (k0 < D_) ? *(const v2f*)(a0 + k0) : *(const v2f*)(a1 + k0 - D_);
    #pragma unroll
    for (int j = 0; j < 4; ++j) {
      const int o = j * 16 + col;
      v2f wr = *(const v2f*)(Wr + (size_t)o * 128 + k0 + kh * 2);
      v2f wz = *(const v2f*)(Wz + (size_t)o * 128 + k0 + kh * 2);
      cr[j] = wmma_f32(a, wr, cr[j]);
      cz[j] = wmma_f32(a, wz, cz[j]);
    }
  }

  // sigmoid; stash z in cz; write r*state to LDS in [row][feature] order.
  #pragma unroll
  for (int j = 0; j < 4; ++j) {
    const int   o   = j * 16 + col;
    const float brv = br[o], bzv = bz[o];
    #pragma unroll
    for (int v = 0; v < 8; ++v) {
      const int   m  = v + kh * 8;
      const float r  = 1.f / (1.f + expf(-(cr[j][v] + brv)));
      const float z  = 1.f / (1.f + expf(-(cz[j][v] + bzv)));
      cz[j][v]       = z;
      const float pv = prop[((size_t)b * N_ + n0 + m) * D_ + o];
      rp[m * 64 + o] = r * pv;
    }
  }
  __syncthreads();

  // --- h_hat pre-activation, K = 128 (second half reads r*s from LDS) ---
  v8f ch[4] = {vzero8(), vzero8(), vzero8(), vzero8()};
  #pragma unroll 2
  for (int k0 = 0; k0 < 2 * D_; k0 += 4) {
    v2f a;
    if (k0 < D_) a = *(const v2f*)(a0 + k0);
    else         a = *(const v2f*)(&rp[col * 64 + (k0 - D_) + kh * 2]);
    #pragma unroll
    for (int j = 0; j < 4; ++j) {
      v2f wh = *(const v2f*)(Wh + (size_t)(j * 16 + col) * 128 + k0 + kh * 2);
      ch[j] = wmma_f32(a, wh, ch[j]);
    }
  }

  #pragma unroll
  for (int j = 0; j < 4; ++j) {
    const int   o   = j * 16 + col;
    const float bhv = bh[o];
    #pragma unroll
    for (int v = 0; v < 8; ++v) {
      const int   m  = v + kh * 8;
      const float hh = tanhf(ch[j][v] + bhv);
      const float z  = cz[j][v];
      const float pv = prop[((size_t)b * N_ + n0 + m) * D_ + o];
      new_state[((size_t)b * N_ + n0 + m) * D_ + o] = (1.f - z) * pv + z * hh;
    }
  }
}

// ---------------------------------------------------------------------------
// Kernel 4: output head. h = tanh([s', ann] Wo1^T + bo1); out = h . Wo2 + bo2
// K = 96 WMMA, then half-wave shfl_xor reduction over the 64 output features.
// ---------------------------------------------------------------------------
__global__ __launch_bounds__(32) void k_out(
    const float* __restrict__ new_state, const float* __restrict__ ann,
    const float* __restrict__ Wo1, const float* __restrict__ bo1,
    const float* __restrict__ Wo2, const float* __restrict__ bo2,
    float* __restrict__ out) {
  const int lane = threadIdx.x & 31;
  const int b    = blockIdx.x >> 8;
  const int n0   = (blockIdx.x & 255) << 4;
  const int col  = lane & 15;
  const int kh   = lane >> 4;

  const float* s0 = new_state + ((size_t)b * N_ + n0 + col) * D_ + kh * 2;
  const float* s1 = ann + ((size_t)b * N_ + n0 + col) * AD_ + kh * 2;

  v8f c[4] = {vzero8(), vzero8(), vzero8(), vzero8()};
  #pragma unroll 2
  for (int k0 = 0; k0 < D_ + AD_; k0 += 4) {
    v2f a = (k0 < D_) ? *(const v2f*)(s0 + k0) : *(const v2f*)(s1 + k0 - D_);
    #pragma unroll
    for (int j = 0; j < 4; ++j) {
      v2f w = *(const v2f*)(Wo1 + (size_t)(j * 16 + col) * (D_ + AD_) + k0 + kh * 2);
      c[j] = wmma_f32(a, w, c[j]);
    }
  }

  float s[8];
  #pragma unroll
  for (int v = 0; v < 8; ++v) s[v] = 0.f;
  #pragma unroll
  for (int j = 0; j < 4; ++j) {
    const int   o  = j * 16 + col;
    const float b1 = bo1[o], w2 = Wo2[o];
    #pragma unroll
    for (int v = 0; v < 8; ++v) s[v] += tanhf(c[j][v] + b1) * w2;
  }
  // Reduce over the 16 feature-columns held by each half-wave.
  #pragma unroll
  for (int v = 0; v < 8; ++v) {
    float x = s[v];
    x += __shfl_xor(x, 1, 32);
    x += __shfl_xor(x, 2, 32);
    x += __shfl_xor(x, 4, 32);
    x += __shfl_xor(x, 8, 32);
    if (col == 0) out[(size_t)b * N_ + n0 + kh * 8 + v] = x + bo2[0];
  }
}

// ---------------------------------------------------------------------------
extern "C" void kernel_launch(void* const* d_in, const int* in_sizes, int n_in,
                              void* d_out, int out_size, void* d_ws, size_t ws_size,
                              hipStream_t stream) {
  const float* prop = (const float*)d_in[0];
  const float* ann  = (const float*)d_in[1];
  const float* Amat = (const float*)d_in[2];
  const float* Win  = (const float*)d_in[3];
  const float* bin  = (const float*)d_in[4];
  const float* Wr   = (const float*)d_in[5];
  const float* br   = (const float*)d_in[6];
  const float* Wz   = (const float*)d_in[7];
  const float* bz   = (const float*)d_in[8];
  const float* Wh   = (const float*)d_in[9];
  const float* bh   = (const float*)d_in[10];
  const float* Wo1  = (const float*)d_in[11];
  const float* bo1  = (const float*)d_in[12];
  const float* Wo2  = (const float*)d_in[13];
  const float* bo2  = (const float*)d_in[14];
  float* out = (float*)d_out;

  // Workspace: 8 MB in_flat_T + 2 MB a_in + 2 MB new_state = 12 MB.
  float* in_flat_T = (float*)d_ws;                            // [B][D][NE]
  float* a_in      = in_flat_T + (size_t)B_ * D_ * NE_;       // [B][N][D]
  float* new_state = a_in + (size_t)B_ * N_ * D_;             // [B][N][D]

  k_in_transform<<<256, 256, 0, stream>>>(prop, Win, bin, in_flat_T);
  k_agg<<<B_ * 128, 256, 0, stream>>>(Amat, in_flat_T, a_in);
  k_gru<<<B_ * 256, 32, 0, stream>>>(a_in, prop, Wr, br, Wz, bz, Wh, bh, new_state);
  k_out<<<B_ * 256, 32, 0, stream>>>(new_state, ann, Wo1, bo1, Wo2, bo2, out);
}